// WeightedLocalCosineSimilarity_22660247453707
// MI455X (gfx1250) — compile-verified
//
#include <hip/hip_runtime.h>

typedef __attribute__((ext_vector_type(2))) float v2f;
typedef __attribute__((ext_vector_type(8))) float v8f;

#define BN 16
#define CN 64
#define HN 256
#define WN 256
#define PLANE (HN * WN)        /* 65536 */
#define EPSV 1e-6f

// ---------------------------------------------------------------------------
// Stage 1: channel reduction.  out planes: [xx(16 imgs)][xy(16)][yy(16)]
// Bandwidth-dominant kernel: 512 MB of reads, float4-vectorized along W,
// coalesced across lanes; channel loop strides by PLANE.
// ---------------------------------------------------------------------------
__global__ void wlcs_stage1_reduce(const float* __restrict__ x,
                                   const float* __restrict__ y,
                                   float* __restrict__ s) {
  int idx = blockIdx.x * blockDim.x + threadIdx.x;   // 0 .. B*PLANE/4 - 1
  int b   = idx >> 14;                               // / (PLANE/4)
  int r4  = idx & 16383;                             // float4 group in plane
  size_t base = (size_t)(b * CN) * PLANE + (size_t)r4 * 4;

  float4 axx = make_float4(0.f, 0.f, 0.f, 0.f);
  float4 axy = make_float4(0.f, 0.f, 0.f, 0.f);
  float4 ayy = make_float4(0.f, 0.f, 0.f, 0.f);

#pragma unroll 4
  for (int c = 0; c < CN; ++c) {
    const float4 xv = *(const float4*)(x + base + (size_t)c * PLANE);
    const float4 yv = *(const float4*)(y + base + (size_t)c * PLANE);
    axx.x += xv.x * xv.x; axx.y += xv.y * xv.y;
    axx.z += xv.z * xv.z; axx.w += xv.w * xv.w;
    axy.x += xv.x * yv.x; axy.y += xv.y * yv.y;
    axy.z += xv.z * yv.z; axy.w += xv.w * yv.w;
    ayy.x += yv.x * yv.x; ayy.y += yv.y * yv.y;
    ayy.z += yv.z * yv.z; ayy.w += yv.w * yv.w;
  }

  size_t o = (size_t)b * PLANE + (size_t)r4 * 4;
  *(float4*)(s + (size_t)0 * BN * PLANE + o) = axx;
  *(float4*)(s + (size_t)1 * BN * PLANE + o) = axy;
  *(float4*)(s + (size_t)2 * BN * PLANE + o) = ayy;
}

// ---------------------------------------------------------------------------
// Stage 2: horizontal 1D Gaussian as banded matmul via V_WMMA_F32_16X16X4_F32.
// One wave computes one 16x16 output tile; 7 chained K=4 WMMAs cover the
// 26-column input footprint (K padded to 28; B=0 kills the pad).
//   A[m][k] = src[h0+m][w0-5+k]   (clamped + masked to zero at image edge)
//   B[k][n] = g1[k-n], 0 <= k-n <= 10, else 0   (Toeplitz band)
// Lane layout per ISA: half = lane>>4, kbase = 4*s + 2*half, v0->kbase,
// v1->kbase+1; D vgpr r -> row h0 + r + 8*half, col w0 + (lane&15).
// B operands are data-independent: built up-front with clamped (non-
// divergent) LDS reads so the WMMA chain only waits on the A-column loads.
// ---------------------------------------------------------------------------
__global__ void wlcs_stage2_hblur_wmma(const float* __restrict__ s,
                                       const float* __restrict__ gauss,
                                       float* __restrict__ t) {
  __shared__ float g1s[11];
  if (threadIdx.x < 11) {
    // rank-1 factor of the 2D Gaussian: u[i] = g[i][5] / sqrt(g[5][5])
    g1s[threadIdx.x] = gauss[threadIdx.x * 11 + 5] * rsqrtf(gauss[60]);
  }
  __syncthreads();

  const int lane   = threadIdx.x & 31;
  const int waveId = (blockIdx.x * blockDim.x + threadIdx.x) >> 5;
  const int img    = waveId >> 8;        // 48 images, 256 tiles each
  const int tile   = waveId & 255;
  const int h0     = (tile >> 4) << 4;
  const int w0     = (tile & 15) << 4;

  const float* __restrict__ src = s + (size_t)img * PLANE;
  float*       __restrict__ dst = t + (size_t)img * PLANE;

  const int m    = lane & 15;   // A row in tile; also B column index N
  const int half = lane >> 4;
  const int row  = h0 + m;

  // --- Precompute the 7 B-operand pairs (band weights), branch-free -------
  float b0a[7], b1a[7];
#pragma unroll
  for (int st = 0; st < 7; ++st) {
    const int kb = st * 4 + half * 2;
    int t0 = kb - m;
    int t1 = kb + 1 - m;
    int t0c = min(max(t0, 0), 10);
    int t1c = min(max(t1, 0), 10);
    float w0v = g1s[t0c];
    float w1v = g1s[t1c];
    b0a[st] = (t0 == t0c) ? w0v : 0.f;
    b1a[st] = (t1 == t1c) ? w1v : 0.f;
  }

  // --- A loads + chained WMMA accumulation --------------------------------
  v8f acc = {0.f, 0.f, 0.f, 0.f, 0.f, 0.f, 0.f, 0.f};

#pragma unroll
  for (int st = 0; st < 7; ++st) {
    const int kb = st * 4 + half * 2;

    int c0  = w0 - 5 + kb;
    int c1  = c0 + 1;
    int c0c = min(max(c0, 0), WN - 1);
    int c1c = min(max(c1, 0), WN - 1);
    float a0 = src[row * WN + c0c];
    float a1 = src[row * WN + c1c];
    a0 = (c0 == c0c) ? a0 : 0.f;
    a1 = (c1 == c1c) ? a1 : 0.f;

    v2f av; av[0] = a0;      av[1] = a1;
    v2f bv; bv[0] = b0a[st]; bv[1] = b1a[st];

    acc = __builtin_amdgcn_wmma_f32_16x16x4_f32(
        /*neg_a=*/false, av, /*neg_b=*/false, bv,
        /*c_mod=*/(short)0, acc, /*reuse_a=*/false, /*reuse_b=*/false);
  }

#pragma unroll
  for (int r = 0; r < 8; ++r) {
    int orow = h0 + r + half * 8;
    dst[orow * WN + w0 + m] = acc[r];
  }
}

// ---------------------------------------------------------------------------
// Stage 3: vertical 1D Gaussian + cosine-similarity combine.
// One thread per output pixel; ~16 MB of traffic, negligible vs stage 1.
// ---------------------------------------------------------------------------
__global__ void wlcs_stage3_vblur_combine(const float* __restrict__ t,
                                          const float* __restrict__ gauss,
                                          float* __restrict__ out) {
  __shared__ float g1s[11];
  if (threadIdx.x < 11) {
    g1s[threadIdx.x] = gauss[threadIdx.x * 11 + 5] * rsqrtf(gauss[60]);
  }
  __syncthreads();

  int idx = blockIdx.x * blockDim.x + threadIdx.x;  // 0 .. B*PLANE - 1
  int b   = idx >> 16;
  int pix = idx & 65535;
  int h   = pix >> 8;
  int w   = pix & 255;

  const float* __restrict__ pxx = t + (size_t)(0 * BN + b) * PLANE;
  const float* __restrict__ pxy = t + (size_t)(1 * BN + b) * PLANE;
  const float* __restrict__ pyy = t + (size_t)(2 * BN + b) * PLANE;

  float sxx = 0.f, sxy = 0.f, syy = 0.f;
#pragma unroll
  for (int tt = 0; tt < 11; ++tt) {
    int hh  = h + tt - 5;
    int hhc = min(max(hh, 0), HN - 1);
    float gw = (hh == hhc) ? g1s[tt] : 0.f;
    int off  = hhc * WN + w;
    sxx += pxx[off] * gw;
    sxy += pxy[off] * gw;
    syy += pyy[off] * gw;
  }
  out[idx] = sxy / (sqrtf(sxx) * sqrtf(syy) + EPSV);
}

// ---------------------------------------------------------------------------
extern "C" void kernel_launch(void* const* d_in, const int* in_sizes, int n_in,
                              void* d_out, int out_size, void* d_ws, size_t ws_size,
                              hipStream_t stream) {
  const float* x     = (const float*)d_in[0];  // (16,64,256,256) f32
  const float* y     = (const float*)d_in[1];  // (16,64,256,256) f32
  const float* gauss = (const float*)d_in[2];  // (11,11) f32
  float*       out   = (float*)d_out;          // (16,256,256) f32

  float* s0 = (float*)d_ws;                      // 3*16*PLANE floats (12 MB)
  float* s1 = s0 + (size_t)3 * BN * PLANE;       // 3*16*PLANE floats (12 MB)

  // Stage 1: B*PLANE/4 threads, float4 along W
  wlcs_stage1_reduce<<<1024, 256, 0, stream>>>(x, y, s0);

  // Stage 2: 48 imgs * 256 tiles = 12288 waves, 8 waves/block
  wlcs_stage2_hblur_wmma<<<1536, 256, 0, stream>>>(s0, gauss, s1);

  // Stage 3: one thread per output pixel
  wlcs_stage3_vblur_combine<<<4096, 256, 0, stream>>>(s1, gauss, out);
}